// Physics_4166118277763
// MI455X (gfx1250) — compile-verified
//
#include <hip/hip_runtime.h>
#include <math.h>

// MI455X (gfx1250) per-particle plasticity projection.
// Async-LDS staged I/O (gfx1250 ASYNCcnt path) + single-instruction hardware
// transcendentals (v_log/v_exp/v_rcp/v_rsq/v_sqrt). Algebraic collapse:
//   out = F * V diag((0.5*sp+0.3*ss)/s) V^T + 0.2*J^(1/3)*I.

#define TPB 256
#define TILE_FLOATS (TPB * 9)          // 2304 floats = 9216 B per block tile
#define TILE_VEC4   (TILE_FLOATS / 4)  // 576 16-byte packets

typedef int int4v __attribute__((ext_vector_type(4)));
typedef __attribute__((address_space(1))) int4v glb_int4;
typedef __attribute__((address_space(3))) int4v lds_int4;

// ---- gfx1250 async global<->LDS paths (probe-confirmed present) ----
#ifdef __has_builtin
#if __has_builtin(__builtin_amdgcn_global_load_async_to_lds_b128)
#define HAVE_ASYNC_LD 1
#endif
#if __has_builtin(__builtin_amdgcn_global_store_async_from_lds_b128)
#define HAVE_ASYNC_ST 1
#endif
#if __has_builtin(__builtin_amdgcn_s_wait_asynccnt)
#define HAVE_WAIT_ASYNC 1
#endif
#if __has_builtin(__builtin_amdgcn_rcpf) && __has_builtin(__builtin_amdgcn_rsqf) && \
    __has_builtin(__builtin_amdgcn_sqrtf) && __has_builtin(__builtin_amdgcn_logf) && \
    __has_builtin(__builtin_amdgcn_exp2f)
#define HAVE_FAST_MATH 1
#endif
#endif

__device__ __forceinline__ void wait_async0() {
#if HAVE_WAIT_ASYNC
  __builtin_amdgcn_s_wait_asynccnt(0);
#else
  asm volatile("s_wait_asynccnt 0" ::: "memory");
#endif
}

// ---- single-instruction hardware math (fallback to libm if absent) ----
#if HAVE_FAST_MATH
__device__ __forceinline__ float frcp(float x)  { return __builtin_amdgcn_rcpf(x); }
__device__ __forceinline__ float frsq(float x)  { return __builtin_amdgcn_rsqf(x); }
__device__ __forceinline__ float fsqrt(float x) { return __builtin_amdgcn_sqrtf(x); }
__device__ __forceinline__ float flog(float x)  { return __builtin_amdgcn_logf(x) * 0.69314718055994531f; }   // ln2 * log2(x)
__device__ __forceinline__ float fexp(float x)  { return __builtin_amdgcn_exp2f(x * 1.44269504088896340f); }  // exp2(x*log2e)
#else
__device__ __forceinline__ float frcp(float x)  { return 1.0f / x; }
__device__ __forceinline__ float frsq(float x)  { return rsqrtf(x); }
__device__ __forceinline__ float fsqrt(float x) { return sqrtf(x); }
__device__ __forceinline__ float flog(float x)  { return logf(x); }
__device__ __forceinline__ float fexp(float x)  { return expf(x); }
#endif

// One Jacobi rotation in plane (p,q), r = remaining index.
// app/aqq/apq/arp/arq reference the 6 unique entries of symmetric A;
// v*p / v*q reference columns p,q of the accumulated eigenvector matrix V.
__device__ __forceinline__ void jrot(float& app, float& aqq, float& apq,
                                     float& arp, float& arq,
                                     float& v0p, float& v0q,
                                     float& v1p, float& v1q,
                                     float& v2p, float& v2q) {
  float denom = 2.0f * apq;
  float theta = (fabsf(denom) > 1e-30f) ? (aqq - app) * frcp(denom) : 3.0e38f;
  float t = copysignf(frcp(fabsf(theta) + fsqrt(fmaf(theta, theta, 1.0f))), theta);
  float c = frsq(fmaf(t, t, 1.0f));
  float s = t * c;
  float tapq = t * apq;
  app -= tapq;
  aqq += tapq;
  apq = 0.0f;
  float x, y;
  x = arp; y = arq; arp = c * x - s * y; arq = s * x + c * y;
  x = v0p; y = v0q; v0p = c * x - s * y; v0q = s * x + c * y;
  x = v1p; y = v1q; v1p = c * x - s * y; v1q = s * x + c * y;
  x = v2p; y = v2q; v2p = c * x - s * y; v2q = s * x + c * y;
}

__global__ __launch_bounds__(TPB)
void plasticity_kernel(const float* __restrict__ Fg,
                       const float* __restrict__ yml_p,
                       const float* __restrict__ nu_p,
                       const float* __restrict__ ys_p,
                       const float* __restrict__ alpha_p,
                       const float* __restrict__ coh_p,
                       float* __restrict__ outg, int n) {
  __shared__ __align__(16) float tile[TILE_FLOATS];

  const int tid = threadIdx.x;
  const int blk = blockIdx.x;
  const int base = blk * TILE_FLOATS;          // float index of tile start
  const int total = n * 9;
  const bool full = (base + TILE_FLOATS) <= total;

  // ---------------- stage in: 9216 B contiguous, async B128 -> LDS ----------------
  if (full) {
#if HAVE_ASYNC_LD
    float* gsrc = const_cast<float*>(Fg) + base;
    for (int i = tid; i < TILE_VEC4; i += TPB) {
      __builtin_amdgcn_global_load_async_to_lds_b128(
          (glb_int4*)(gsrc + i * 4),
          (lds_int4*)(tile + i * 4),
          0, 0);
    }
    wait_async0();
#else
    const float4* g4 = (const float4*)(Fg + base);
    float4* l4 = (float4*)tile;
    for (int i = tid; i < TILE_VEC4; i += TPB) l4[i] = g4[i];
#endif
  } else {
    for (int i = tid; i < TILE_FLOATS; i += TPB) {
      int g = base + i;
      tile[i] = (g < total) ? Fg[g] : 0.0f;
    }
  }
  __syncthreads();

  // ---------------- per-particle compute (registers only) ----------------
  const float* f = &tile[tid * 9];   // stride-9 LDS reads: 9 coprime 64 -> no bank conflicts
  float f00 = f[0], f01 = f[1], f02 = f[2];
  float f10 = f[3], f11 = f[4], f12 = f[5];
  float f20 = f[6], f21 = f[7], f22 = f[8];

  // uniform material constants
  const float E  = fexp(yml_p[0]);
  const float nu = nu_p[0];
  const float mu = E * 0.5f * frcp(1.0f + nu);
  const float la = E * nu * frcp((1.0f + nu) * (1.0f - 2.0f * nu));
  const float rcp2mu = frcp(2.0f * mu);
  const float Ydiv  = ys_p[0] * rcp2mu;                       // Y/(2mu)
  const float dpco  = (3.0f * la + 2.0f * mu) * rcp2mu * alpha_p[0];
  const float coh = coh_p[0];

  // A = F^T F (symmetric, 6 uniques)
  float a00 = f00*f00 + f10*f10 + f20*f20;
  float a11 = f01*f01 + f11*f11 + f21*f21;
  float a22 = f02*f02 + f12*f12 + f22*f22;
  float a01 = f00*f01 + f10*f11 + f20*f21;
  float a02 = f00*f02 + f10*f12 + f20*f22;
  float a12 = f01*f02 + f11*f12 + f21*f22;

  float v00 = 1.f, v01 = 0.f, v02 = 0.f;
  float v10 = 0.f, v11 = 1.f, v12 = 0.f;
  float v20 = 0.f, v21 = 0.f, v22 = 1.f;

  // 3 cyclic sweeps reach fp32 roundoff for these well-conditioned Gram
  // matrices; near-degenerate eigenpairs don't need accurate V (w's collapse).
#pragma unroll
  for (int sw = 0; sw < 3; ++sw) {
    jrot(a00, a11, a01, a02, a12, v00, v01, v10, v11, v20, v21); // (p,q)=(0,1)
    jrot(a00, a22, a02, a01, a12, v00, v02, v10, v12, v20, v22); // (p,q)=(0,2)
    jrot(a11, a22, a12, a01, a02, v01, v02, v11, v12, v21, v22); // (p,q)=(1,2)
  }

  float s0 = fsqrt(fmaxf(a00, 0.f));
  float s1 = fsqrt(fmaxf(a11, 0.f));
  float s2 = fsqrt(fmaxf(a22, 0.f));

  // ---- plasticine (von Mises) ----
  float sp0 = fmaxf(s0, 0.01f), sp1 = fmaxf(s1, 0.01f), sp2 = fmaxf(s2, 0.01f);
  float e0 = flog(sp0), e1 = flog(sp1), e2 = flog(sp2);
  float mp = (e0 + e1 + e2) * (1.0f / 3.0f);
  float b0 = e0 - mp, b1 = e1 - mp, b2 = e2 - mp;
  float bn = fsqrt(b0*b0 + b1*b1 + b2*b2) + 1e-5f;
  float dgam = bn - Ydiv;
  float kf = dgam * frcp(bn);
  bool py = (dgam > 0.f);
  float pl0 = py ? fexp(e0 - kf * b0) : sp0;
  float pl1 = py ? fexp(e1 - kf * b1) : sp1;
  float pl2 = py ? fexp(e2 - kf * b2) : sp2;

  // ---- sand (Drucker-Prager) ----
  // log(max(s,0.05)) == max(log(max(s,0.01)), log(0.05)) (log monotone, s>=0)
  const float LOG005 = -2.9957322735539909f;
  float g0 = fmaxf(e0, LOG005), g1 = fmaxf(e1, LOG005), g2 = fmaxf(e2, LOG005);
  float trs = g0 + g1 + g2;
  float ms = trs * (1.0f / 3.0f);
  float h0 = g0 - ms, h1 = g1 - ms, h2 = g2 - ms;
  float hn = fsqrt(h0*h0 + h1*h1 + h2*h2);
  float shifted = trs - coh * 3.0f;
  float dg = hn + shifted * dpco;
  float q = fmaxf(dg, 0.f) * frcp(fmaxf(hn, 1e-30f));
  bool cy = (shifted < 0.f);
  float sd0 = fexp(cy ? (g0 - q * h0) : coh);
  float sd1 = fexp(cy ? (g1 - q * h1) : coh);
  float sd2 = fexp(cy ? (g2 - q * h2) : coh);

  // ---- water: 0.2 * det(F)^(1/3) ----
  float J = f00 * (f11 * f22 - f12 * f21)
          - f01 * (f10 * f22 - f12 * f20)
          + f02 * (f10 * f21 - f11 * f20);
  float wtr = 0.2f * fexp(flog(J) * (1.0f / 3.0f));  // NaN for J<0, same as jnp.power

  // ---- collapse: out = F * (V diag(w) V^T) + wtr*I,  w = (0.5 pl + 0.3 sd)/s ----
  float w0 = (0.5f * pl0 + 0.3f * sd0) * frcp(fmaxf(s0, 1e-12f));
  float w1 = (0.5f * pl1 + 0.3f * sd1) * frcp(fmaxf(s1, 1e-12f));
  float w2 = (0.5f * pl2 + 0.3f * sd2) * frcp(fmaxf(s2, 1e-12f));

  float t00 = w0*v00, t01 = w1*v01, t02 = w2*v02;
  float t10 = w0*v10, t11 = w1*v11, t12 = w2*v12;
  float t20 = w0*v20, t21 = w1*v21, t22 = w2*v22;

  float m00 = t00*v00 + t01*v01 + t02*v02;
  float m01 = t00*v10 + t01*v11 + t02*v12;
  float m02 = t00*v20 + t01*v21 + t02*v22;
  float m11 = t10*v10 + t11*v11 + t12*v12;
  float m12 = t10*v20 + t11*v21 + t12*v22;
  float m22 = t20*v20 + t21*v21 + t22*v22;

  float o00 = f00*m00 + f01*m01 + f02*m02 + wtr;
  float o01 = f00*m01 + f01*m11 + f02*m12;
  float o02 = f00*m02 + f01*m12 + f02*m22;
  float o10 = f10*m00 + f11*m01 + f12*m02;
  float o11 = f10*m01 + f11*m11 + f12*m12 + wtr;
  float o12 = f10*m02 + f11*m12 + f12*m22;
  float o20 = f20*m00 + f21*m01 + f22*m02;
  float o21 = f20*m01 + f21*m11 + f22*m12;
  float o22 = f20*m02 + f21*m12 + f22*m22 + wtr;

  __syncthreads();   // everyone done reading the input tile

  float* o = &tile[tid * 9];
  o[0] = o00; o[1] = o01; o[2] = o02;
  o[3] = o10; o[4] = o11; o[5] = o12;
  o[6] = o20; o[7] = o21; o[8] = o22;

  __syncthreads();   // results staged

  // ---------------- stage out: async B128 LDS -> global ----------------
  if (full) {
#if HAVE_ASYNC_ST
    float* gdst = outg + base;
    for (int i = tid; i < TILE_VEC4; i += TPB) {
      __builtin_amdgcn_global_store_async_from_lds_b128(
          (glb_int4*)(gdst + i * 4),
          (lds_int4*)(tile + i * 4),
          0, 0);
    }
    wait_async0();   // harmless; S_ENDPGM also implies wait-idle
#else
    float4* g4 = (float4*)(outg + base);
    const float4* l4 = (const float4*)tile;
    for (int i = tid; i < TILE_VEC4; i += TPB) g4[i] = l4[i];
#endif
  } else {
    for (int i = tid; i < TILE_FLOATS; i += TPB) {
      int g = base + i;
      if (g < total) outg[g] = tile[i];
    }
  }
}

extern "C" void kernel_launch(void* const* d_in, const int* in_sizes, int n_in,
                              void* d_out, int out_size, void* d_ws, size_t ws_size,
                              hipStream_t stream) {
  const float* F     = (const float*)d_in[0];
  const float* yml   = (const float*)d_in[1];
  const float* nu    = (const float*)d_in[2];
  const float* ys    = (const float*)d_in[3];
  const float* alpha = (const float*)d_in[4];
  const float* coh   = (const float*)d_in[5];
  float* out = (float*)d_out;

  int n = in_sizes[0] / 9;                  // number of particles
  int blocks = (n + TPB - 1) / TPB;
  plasticity_kernel<<<blocks, TPB, 0, stream>>>(F, yml, nu, ys, alpha, coh, out, n);
}